// GINModel_17411797418333
// MI455X (gfx1250) — compile-verified
//
#include <hip/hip_runtime.h>
#include <math.h>

typedef float v2f __attribute__((ext_vector_type(2)));
typedef float v8f __attribute__((ext_vector_type(8)));

#define LDH 132   // padded K-stride for 129-wide tiles (129->132, mult of 4; 132%64=4 -> no LDS bank clash)
#define LDC 260   // padded K-stride for 257-wide comb tiles

// ---------------- weight padding (zero-fill K remainder rows) ----------------
__global__ void pad_weights_kernel(const float* __restrict__ W1,
                                   const float* __restrict__ P1,
                                   float* __restrict__ W1p,
                                   float* __restrict__ P1p) {
  int stride = gridDim.x * blockDim.x;
  int idx = blockIdx.x * blockDim.x + threadIdx.x;
  for (int i = idx; i < 132 * 128; i += stride)
    W1p[i] = (i < 129 * 128) ? W1[i] : 0.0f;
  for (int i = idx; i < 260 * 128; i += stride)
    P1p[i] = (i < 257 * 128) ? P1[i] : 0.0f;
}

// ---------------- hbuf = xt (self term of GIN agg), zero t_pred ----------------
__global__ void init_hbuf_kernel(const float* __restrict__ x,
                                 const float* __restrict__ t,
                                 float* __restrict__ hbuf,
                                 float* __restrict__ tpred, int N) {
  int i = blockIdx.x;
  if (i >= N) return;
  for (int c = threadIdx.x; c < LDH; c += blockDim.x) {
    float v = 0.0f;
    if (c < 128)      v = x[i * 128 + c];
    else if (c == 128) v = t[i];
    hbuf[i * LDH + c] = v;
  }
  if (threadIdx.x == 0) tpred[i] = 0.0f;
}

// ---------------- edge scatter-add: one wave32 per edge ----------------
__global__ void scatter_edges_kernel(const long long* __restrict__ ei,
                                     const float* __restrict__ x,
                                     const float* __restrict__ t,
                                     float* __restrict__ hbuf, int E) {
  int wid  = (int)((blockIdx.x * blockDim.x + threadIdx.x) >> 5);
  int lane = threadIdx.x & 31;
  if (wid >= E) return;
  int s = (int)ei[wid];                 // edge_index[0][e]
  int d = (int)ei[(long long)E + wid];  // edge_index[1][e]
  for (int f = lane; f < 129; f += 32) {
    float v = (f < 128) ? x[s * 128 + f] : t[s];
    atomicAdd(&hbuf[d * LDH + f], v);   // global_atomic_add_f32, L2-resident
  }
}

// ---------------- fp32 WMMA helpers (V_WMMA_F32_16X16X4_F32) ----------------
// A 16x4 f32 layout: lanes 0-15 -> M=lane, VGPR0=K0,VGPR1=K1 ; lanes 16-31 -> K2,K3
// B 4x16  f32 layout: lanes 0-15 -> N=lane, VGPR0=K0,VGPR1=K1 ; lanes 16-31 -> K2,K3
template <int K>
__device__ __forceinline__ v8f wmma_chain(const float* __restrict__ A, int lda,
                                          const float* __restrict__ B, int ldb,
                                          int r, int halfk, int n, v8f acc) {
#pragma unroll
  for (int k0 = 0; k0 < K; k0 += 4) {
    int kk = k0 + 2 * halfk;
    v2f a = *(const v2f*)(A + r * lda + kk);     // 8B aligned (kk even, lda even)
    v2f b;
    b.x = B[kk * ldb + n];
    b.y = B[(kk + 1) * ldb + n];
    acc = __builtin_amdgcn_wmma_f32_16x16x4_f32(false, a, false, b,
                                                (short)0, acc, false, false);
  }
  return acc;
}

// C/D 16x16 f32 layout: VGPR v, lanes 0-15 -> (M=v, N=lane); lanes 16-31 -> (M=v+8, N=lane-16)
__device__ __forceinline__ void store_frag(float* __restrict__ D, int ldd,
                                           int halfk, int n, v8f d) {
#pragma unroll
  for (int v = 0; v < 8; ++v) D[(v + 8 * halfk) * ldd + n] = d[v];
}

// ---------------- fused 5-layer MLP: one block per 16 nodes, 8 waves x 16 cols ----------------
__global__ void __launch_bounds__(256)
fused_mlp_kernel(const float* __restrict__ hbuf,
                 const float* __restrict__ x,  const float* __restrict__ t,
                 const float* __restrict__ W1p, const float* __restrict__ b1,
                 const float* __restrict__ W2,  const float* __restrict__ b2,
                 const float* __restrict__ P1p, const float* __restrict__ pb1,
                 const float* __restrict__ P2,  const float* __restrict__ pb2,
                 const float* __restrict__ P3,  const float* __restrict__ pb3,
                 float* __restrict__ out_y, int N) {
  __shared__ float bufA[16 * LDH];
  __shared__ float bufH[16 * LDH];
  __shared__ float comb[16 * LDC];

  const int tid  = threadIdx.x;
  const int lane = tid & 31;
  const int wave = tid >> 5;
  const int m0   = blockIdx.x * 16;

  // load hsum tile [16x132] and xt tile into comb cols 128..259 (257..259 zero)
  for (int i = tid; i < 16 * LDH; i += 256) {
    int rr = i / LDH, cc = i % LDH;
    int g = m0 + rr;
    bufA[i] = (g < N) ? hbuf[g * LDH + cc] : 0.0f;
    float v = 0.0f;
    if (g < N) {
      if (cc < 128)      v = x[g * 128 + cc];
      else if (cc == 128) v = t[g];
    }
    comb[rr * LDC + 128 + cc] = v;
  }
  __syncthreads();

  const int halfk = lane >> 4;
  const int r     = lane & 15;
  const int n     = wave * 16 + r;

  // Stage 1: relu(hsum @ W1 + b1)  [K=132 padded]
  v8f acc = {0.f, 0.f, 0.f, 0.f, 0.f, 0.f, 0.f, 0.f};
  acc = wmma_chain<132>(bufA, LDH, W1p, 128, r, halfk, n, acc);
  {
    float bias = b1[n];
#pragma unroll
    for (int v = 0; v < 8; ++v) { float u = acc[v] + bias; acc[v] = u > 0.f ? u : 0.f; }
  }
  store_frag(bufH, LDH, halfk, n, acc);
  __syncthreads();

  // Stage 2: relu(tanh(h1 @ W2 + b2)) -> comb cols 0..127
  acc = (v8f){0.f, 0.f, 0.f, 0.f, 0.f, 0.f, 0.f, 0.f};
  acc = wmma_chain<128>(bufH, LDH, W2, 128, r, halfk, n, acc);
  {
    float bias = b2[n];
#pragma unroll
    for (int v = 0; v < 8; ++v) { float u = tanhf(acc[v] + bias); acc[v] = u > 0.f ? u : 0.f; }
  }
  store_frag(comb, LDC, halfk, n, acc);
  __syncthreads();

  // Stage 3: leaky_relu(comb @ P1 + pb1, 0.2)  [K=260 padded]
  acc = (v8f){0.f, 0.f, 0.f, 0.f, 0.f, 0.f, 0.f, 0.f};
  acc = wmma_chain<260>(comb, LDC, P1p, 128, r, halfk, n, acc);
  {
    float bias = pb1[n];
#pragma unroll
    for (int v = 0; v < 8; ++v) { float u = acc[v] + bias; acc[v] = u > 0.f ? u : 0.2f * u; }
  }
  store_frag(bufH, LDH, halfk, n, acc);
  __syncthreads();

  // Stage 4: leaky_relu(y1 @ P2 + pb2, 0.2)
  acc = (v8f){0.f, 0.f, 0.f, 0.f, 0.f, 0.f, 0.f, 0.f};
  acc = wmma_chain<128>(bufH, LDH, P2, 128, r, halfk, n, acc);
  {
    float bias = pb2[n];
#pragma unroll
    for (int v = 0; v < 8; ++v) { float u = acc[v] + bias; acc[v] = u > 0.f ? u : 0.2f * u; }
  }
  store_frag(bufA, LDH, halfk, n, acc);
  __syncthreads();

  // Stage 5: y = y2 @ P3 + pb3  (128->1 matvec, 16 rows)
  if (tid < 16) {
    int g = m0 + tid;
    if (g < N) {
      float s = pb3[0];
#pragma unroll 8
      for (int k = 0; k < 128; ++k) s += bufA[tid * LDH + k] * P3[k];
      out_y[g] = s;
    }
  }
}

// ---------------- launch ----------------
extern "C" void kernel_launch(void* const* d_in, const int* in_sizes, int n_in,
                              void* d_out, int out_size, void* d_ws, size_t ws_size,
                              hipStream_t stream) {
  (void)n_in; (void)out_size; (void)ws_size;
  const float*     x   = (const float*)d_in[0];
  const float*     t   = (const float*)d_in[1];
  /* d_in[2] = z, unused by reference */
  const long long* ei  = (const long long*)d_in[3];   // int64 edge_index [2,E]
  const float*     W1  = (const float*)d_in[4];
  const float*     b1  = (const float*)d_in[5];
  const float*     W2  = (const float*)d_in[6];
  const float*     b2  = (const float*)d_in[7];
  const float*     P1  = (const float*)d_in[8];
  const float*     pb1 = (const float*)d_in[9];
  const float*     P2  = (const float*)d_in[10];
  const float*     pb2 = (const float*)d_in[11];
  const float*     P3  = (const float*)d_in[12];
  const float*     pb3 = (const float*)d_in[13];

  const int N = in_sizes[1];        // t: [N]
  const int E = in_sizes[3] / 2;    // edge_index: [2,E]

  // workspace layout
  char*  ws   = (char*)d_ws;
  float* hbuf = (float*)ws;                                   // N * 132 f32
  size_t off  = (size_t)N * LDH * sizeof(float);
  off = (off + 255) & ~(size_t)255;
  float* W1p = (float*)(ws + off);                            // 132*128 f32
  off += (size_t)132 * 128 * sizeof(float);
  off = (off + 255) & ~(size_t)255;
  float* P1p = (float*)(ws + off);                            // 260*128 f32

  float* tpred = (float*)d_out;        // first N outputs (zeros)
  float* out_y = (float*)d_out + N;    // next N outputs

  pad_weights_kernel<<<64, 256, 0, stream>>>(W1, P1, W1p, P1p);
  init_hbuf_kernel<<<N, 128, 0, stream>>>(x, t, hbuf, tpred, N);
  const int epb = 256 / 32;            // 8 edges (waves) per block
  scatter_edges_kernel<<<(E + epb - 1) / epb, 256, 0, stream>>>(ei, x, t, hbuf, E);
  fused_mlp_kernel<<<(N + 15) / 16, 256, 0, stream>>>(hbuf, x, t, W1p, b1, W2, b2,
                                                      P1p, pb1, P2, pb2, P3, pb3,
                                                      out_y, N);
}